// Attention_44195213476226
// MI455X (gfx1250) — compile-verified
//
#include <hip/hip_runtime.h>
#include <hip/hip_bf16.h>

typedef __attribute__((ext_vector_type(2))) float v2f;
typedef __attribute__((ext_vector_type(8))) float v8f;

#define DD 1024
#define SS 1024
#define BB 32
#define NCHUNK 8   // S split for context partials
#define CHUNK  (SS / NCHUNK)

// ---------------------------------------------------------------------------
// Kernel 1: u = Wh^T v and r = Ws^T v via V_WMMA_F32_16X16X4_F32.
// One wave (32 threads) per 16-wide output tile. blockIdx [0,64): u-tiles of Wh,
// [64,128): r-tiles of Ws. A-matrix holds v in row M=0 only (lane0: K0,K1;
// lane16: K2,K3 per ISA 16x4 f32 layout). Branch-free fragment build: every
// lane loads v[k + (lane>>4)*2] and masks with a select, so EXEC stays all-1s
// and the loop body is pure load+cndmask+wmma (no per-iteration exec juggling).
// ---------------------------------------------------------------------------
__global__ __launch_bounds__(32) void uvec_wmma_kernel(
    const float* __restrict__ Wh, const float* __restrict__ Ws,
    const float* __restrict__ v, float* __restrict__ u, float* __restrict__ r)
{
    const int lane = threadIdx.x;
    const int tile = blockIdx.x & 63;
    const int sel  = blockIdx.x >> 6;
    const float* __restrict__ M = sel ? Ws : Wh;
    float* __restrict__ outp    = sel ? r  : u;
    const int dt     = tile * 16;
    const int n      = lane & 15;
    const int lhalf  = lane >> 4;            // 0 for lanes 0-15, 1 for 16-31
    const int krow   = lhalf * 2;            // B rows K+0/K+1 vs K+2/K+3
    const bool amask = (n == 0);             // only lanes 0 and 16 carry A data

    v8f c = {};
#pragma unroll 4
    for (int k = 0; k < DD; k += 4) {
        // A fragment: row M=0 = v[k..k+3]; all other rows zero.
        const float2 av = *(const float2*)(v + k + krow);
        v2f a;
        a[0] = amask ? av.x : 0.0f;
        a[1] = amask ? av.y : 0.0f;

        // B fragment: B[kk, n] = M[(k+kk)*DD + dt + n].
        v2f b;
        b[0] = M[(size_t)(k + krow)     * DD + dt + n];
        b[1] = M[(size_t)(k + krow + 1) * DD + dt + n];

        c = __builtin_amdgcn_wmma_f32_16x16x4_f32(
                false, a, false, b, (short)0, c, false, false);
    }
    if (lane < 16) outp[dt + lane] = c[0];   // C row 0: VGPR0, lanes 0-15
}

// ---------------------------------------------------------------------------
// Kernel 2: tiny dot products. Blocks 0..31: c_b = r . s_t[b]
// (s_t = concat(h_dec, c_dec)). Block 32: vb = v . Ws_b. Block 33: k = v . Wc.
// ---------------------------------------------------------------------------
__global__ __launch_bounds__(128) void small_dots_kernel(
    const float* __restrict__ h_dec, const float* __restrict__ c_dec,
    const float* __restrict__ Wsb,   const float* __restrict__ Wc,
    const float* __restrict__ v,     const float* __restrict__ r,
    float* __restrict__ out)
{
    __shared__ float red[128];
    const int t = threadIdx.x;
    const int b = blockIdx.x;
    float acc = 0.0f;
    if (b < BB) {
        for (int i = t; i < DD; i += 128) {
            float s = (i < 512) ? h_dec[b * 512 + i] : c_dec[b * 512 + (i - 512)];
            acc += r[i] * s;
        }
    } else if (b == BB) {
        for (int i = t; i < DD; i += 128) acc += v[i] * Wsb[i];
    } else {
        for (int i = t; i < DD; i += 128) acc += v[i] * Wc[i];
    }
    red[t] = acc; __syncthreads();
    for (int off = 64; off > 0; off >>= 1) {
        if (t < off) red[t] += red[t + off];
        __syncthreads();
    }
    if (t == 0) out[b] = red[0];
}

// ---------------------------------------------------------------------------
// Kernel 3: scores. One wave32 per (b,s) row: interleaved float4 dot of
// enc[b,s,:] with u, then shuffle reduction. Streams the 128 MB tensor once.
// ---------------------------------------------------------------------------
__global__ __launch_bounds__(256) void score_kernel(
    const float* __restrict__ enc, const float* __restrict__ cov,
    const float* __restrict__ u,   const float* __restrict__ cbv,
    float* __restrict__ scores)
{
    const int lane = threadIdx.x & 31;
    const int wid  = threadIdx.x >> 5;
    const int row  = blockIdx.x * 8 + wid;           // [0, B*S)

    const float4* __restrict__ e4 = (const float4*)(enc + (size_t)row * DD);
    const float4* __restrict__ u4 = (const float4*)u;

    float acc = 0.0f;
#pragma unroll
    for (int j = 0; j < 8; ++j) {
        float4 ev = e4[lane + j * 32];
        float4 uv = u4[lane + j * 32];
        acc += ev.x * uv.x + ev.y * uv.y + ev.z * uv.z + ev.w * uv.w;
    }
#pragma unroll
    for (int off = 16; off > 0; off >>= 1) acc += __shfl_down(acc, off, 32);

    if (lane == 0) {
        const int b = row >> 10;
        scores[row] = acc + cbv[b] + cbv[32] + cbv[33] * cov[row];
    }
}

// ---------------------------------------------------------------------------
// Kernel 4: per-batch softmax + mask + renormalize; emits w and coverage_new.
// ---------------------------------------------------------------------------
__global__ __launch_bounds__(256) void softmax_kernel(
    const float* __restrict__ scores, const float* __restrict__ mask,
    const float* __restrict__ cov, float* __restrict__ w_out,
    float* __restrict__ cov_out)
{
    __shared__ float red[256];
    const int b = blockIdx.x;
    const int t = threadIdx.x;
    const float* __restrict__ sc = scores + b * SS;

    float loc[4];
    float m = -3.402823466e38f;
#pragma unroll
    for (int j = 0; j < 4; ++j) { loc[j] = sc[t + j * 256]; m = fmaxf(m, loc[j]); }
    red[t] = m; __syncthreads();
    for (int off = 128; off > 0; off >>= 1) {
        if (t < off) red[t] = fmaxf(red[t], red[t + off]);
        __syncthreads();
    }
    const float gmax = red[0]; __syncthreads();

    float ex[4]; float se = 0.0f;
#pragma unroll
    for (int j = 0; j < 4; ++j) { ex[j] = expf(loc[j] - gmax); se += ex[j]; }
    red[t] = se; __syncthreads();
    for (int off = 128; off > 0; off >>= 1) {
        if (t < off) red[t] += red[t + off];
        __syncthreads();
    }
    const float gsum = red[0]; __syncthreads();

    float wm[4]; float sw = 0.0f;
#pragma unroll
    for (int j = 0; j < 4; ++j) {
        wm[j] = (ex[j] / gsum) * mask[b * SS + t + j * 256];
        sw += wm[j];
    }
    red[t] = sw; __syncthreads();
    for (int off = 128; off > 0; off >>= 1) {
        if (t < off) red[t] += red[t + off];
        __syncthreads();
    }
    const float gsw = red[0];

#pragma unroll
    for (int j = 0; j < 4; ++j) {
        const int idx = b * SS + t + j * 256;
        const float wv = wm[j] / gsw;
        w_out[idx]   = wv;
        cov_out[idx] = cov[idx] + wv;
    }
}

// ---------------------------------------------------------------------------
// Kernel 5: context partials. grid = (d-tile[4], s-chunk[8], b[32]); threads
// own consecutive d (coalesced 128 B per wave iteration); second pass over enc
// should be mostly L2-resident (128 MB < 192 MB L2).
// ---------------------------------------------------------------------------
__global__ __launch_bounds__(256) void context_partial_kernel(
    const float* __restrict__ enc, const float* __restrict__ w,
    float* __restrict__ partial)
{
    const int d     = blockIdx.x * 256 + threadIdx.x;
    const int chunk = blockIdx.y;
    const int b     = blockIdx.z;

    const float* __restrict__ wrow  = w + b * SS + chunk * CHUNK;
    const float* __restrict__ ebase = enc + ((size_t)(b * SS + chunk * CHUNK)) * DD + d;

    float acc = 0.0f;
    for (int s = 0; s < CHUNK; ++s) acc += wrow[s] * ebase[(size_t)s * DD];
    partial[((b * NCHUNK) + chunk) * DD + d] = acc;
}

// ---------------------------------------------------------------------------
// Kernel 6: reduce partials -> context (B x D).
// ---------------------------------------------------------------------------
__global__ __launch_bounds__(256) void context_reduce_kernel(
    const float* __restrict__ partial, float* __restrict__ ctx)
{
    const int idx = blockIdx.x * 256 + threadIdx.x;   // [0, B*D)
    const int b = idx >> 10;
    const int d = idx & (DD - 1);
    float acc = 0.0f;
#pragma unroll
    for (int c = 0; c < NCHUNK; ++c) acc += partial[((b * NCHUNK) + c) * DD + d];
    ctx[idx] = acc;
}

// ---------------------------------------------------------------------------
extern "C" void kernel_launch(void* const* d_in, const int* in_sizes, int n_in,
                              void* d_out, int out_size, void* d_ws, size_t ws_size,
                              hipStream_t stream) {
    const float* h_dec = (const float*)d_in[0];   // (1,B,512)
    const float* c_dec = (const float*)d_in[1];   // (1,B,512)
    const float* enc   = (const float*)d_in[2];   // (B,S,D)
    const float* mask  = (const float*)d_in[3];   // (B,S)
    const float* cov   = (const float*)d_in[4];   // (B,S)
    const float* Wh    = (const float*)d_in[5];   // (D,D)
    const float* Ws    = (const float*)d_in[6];   // (D,D)
    const float* Wsb   = (const float*)d_in[7];   // (D,)
    const float* Wc    = (const float*)d_in[8];   // (D,1)
    const float* v     = (const float*)d_in[9];   // (1,D)

    float* out     = (float*)d_out;
    float* ctx_out = out;                 // context  (B*D)
    float* w_out   = out + BB * DD;       // w        (B*S)
    float* cov_out = out + 2 * BB * DD;   // coverage (B*S)

    float* ws      = (float*)d_ws;
    float* u       = ws;                  // 1024
    float* r       = ws + 1024;           // 1024
    float* cbv     = ws + 2048;           // 34: c_b[0..31], vb, kconst
    float* scores  = ws + 4096;           // 32768
    float* partial = ws + 36864;          // B*NCHUNK*D = 262144

    uvec_wmma_kernel<<<128, 32, 0, stream>>>(Wh, Ws, v, u, r);
    small_dots_kernel<<<BB + 2, 128, 0, stream>>>(h_dec, c_dec, Wsb, Wc, v, r, cbv);
    score_kernel<<<(BB * SS) / 8, 256, 0, stream>>>(enc, cov, u, cbv, scores);
    softmax_kernel<<<BB, 256, 0, stream>>>(scores, mask, cov, w_out, cov_out);
    context_partial_kernel<<<dim3(DD / 256, NCHUNK, BB), 256, 0, stream>>>(enc, w_out, partial);
    context_reduce_kernel<<<(BB * DD) / 256, 256, 0, stream>>>(partial, ctx_out);
}